// NonLocalBlock2D_37915971289398
// MI455X (gfx1250) — compile-verified
//
#include <hip/hip_runtime.h>
#include <hip/hip_bf16.h>
#include <math.h>

typedef float v2f __attribute__((ext_vector_type(2)));
typedef float v8f __attribute__((ext_vector_type(8)));

static constexpr int NN = 4096;   // H*W
static constexpr int CC = 64;     // channels
static constexpr int C2 = 32;     // bottleneck channels
static constexpr int NB = 8;      // batch

// swap lanes q <-> q+16 (SWAPX16 via ds_swizzle group-of-32: xor=0x10,and=0x1f)
__device__ __forceinline__ float sw16(float v) {
  return __int_as_float(__builtin_amdgcn_ds_swizzle(__float_as_int(v), 0x401f));
}

__device__ __forceinline__ v8f wmma4(v2f a, v2f b, v8f c) {
  // D = A(16x4,f32) * B(4x16,f32) + C(16x16,f32)
  return __builtin_amdgcn_wmma_f32_16x16x4_f32(false, a, false, b, (short)0, c,
                                               false, false);
}

// Convert one 16x16 f32 C/D-layout tile (8 VGPRs) into four K=4 B-fragments.
// C layout: VGPR r holds (M=r, N=lane) for lanes 0-15 and (M=r+8, N=lane-16)
// for lanes 16-31.  B layout (4x16): VGPR0 = {row k0 | row k0+2},
// VGPR1 = {row k0+1 | row k0+3}.
__device__ __forceinline__ void c2b(const v8f& R, bool lo, v2f* B) {
  float s0 = sw16(R[0]), s1 = sw16(R[1]), s2 = sw16(R[2]), s3 = sw16(R[3]);
  float s4 = sw16(R[4]), s5 = sw16(R[5]), s6 = sw16(R[6]), s7 = sw16(R[7]);
  B[0][0] = lo ? R[0] : s2;  B[0][1] = lo ? R[1] : s3;   // rows 0..3
  B[1][0] = lo ? R[4] : s6;  B[1][1] = lo ? R[5] : s7;   // rows 4..7
  B[2][0] = lo ? s0 : R[2];  B[2][1] = lo ? s1 : R[3];   // rows 8..11
  B[3][0] = lo ? s4 : R[6];  B[3][1] = lo ? s5 : R[7];   // rows 12..15
}

// Kernel 1: V[b][n][o] = sum_c g_w[o][c]*x[b][c][n] + g_b[o]   (tiny GEMM)
__global__ __launch_bounds__(256) void gx_kernel(
    const float* __restrict__ x, const float* __restrict__ g_w,
    const float* __restrict__ g_b, float* __restrict__ Vg) {
  int idx = blockIdx.x * 256 + threadIdx.x;       // 0 .. 8*4096-1
  int b = idx >> 12;
  int n = idx & (NN - 1);
  const float* xb = x + (size_t)b * (CC * NN) + n;
  float acc[C2];
#pragma unroll
  for (int o = 0; o < C2; ++o) acc[o] = g_b[o];
  for (int c = 0; c < CC; ++c) {
    float xv = xb[(size_t)c * NN];
#pragma unroll
    for (int o = 0; o < C2; ++o) acc[o] = fmaf(g_w[o * CC + c], xv, acc[o]);
  }
  float* vp = Vg + (size_t)idx * C2;
#pragma unroll
  for (int o = 0; o < C2; ++o) vp[o] = acc[o];
}

// Kernel 2: fused flash attention + output projection + residual.
// Block = 256 threads = 8 waves; each wave owns a 16-query tile (128/block).
__global__ __launch_bounds__(256) void attn_kernel(
    const float* __restrict__ x, const float* __restrict__ Vg,
    const float* __restrict__ W_w, const float* __restrict__ W_b,
    float* __restrict__ out) {
  __shared__ float ldsK[16 * 66];   // K tile [m][c], stride 66 (bank-safe, 8B-aligned pairs)
  __shared__ float ldsV[32 * 18];   // V tile [o][m], stride 18

  const int b    = blockIdx.x >> 5;
  const int qblk = blockIdx.x & 31;
  const int tid  = threadIdx.x;
  const int wave = tid >> 5;
  const int lane = tid & 31;
  const int lq   = lane & 15;
  const bool lo  = lane < 16;
  const int n0   = qblk * 128 + wave * 16;

  const float* xb = x  + (size_t)b * (CC * NN);
  const float* vb = Vg + (size_t)b * (NN * C2);

  // Q^T B-fragments: qb[ck] covers K-rows (channels) 4ck..4ck+3, cols = queries
  v2f qb[16];
  {
    const int crow = lo ? 0 : 2;
#pragma unroll
    for (int ck = 0; ck < 16; ++ck) {
      qb[ck][0] = xb[(size_t)(4 * ck + crow) * NN + n0 + lq];
      qb[ck][1] = xb[(size_t)(4 * ck + crow + 1) * NN + n0 + lq];
    }
  }

  v8f acc0 = {};            // y^T tile, o = 0..15   (N-lane = query)
  v8f acc1 = {};            // y^T tile, o = 16..31
  float m_run = -INFINITY;  // per-query running max (mirrored in both halves)
  float l_run = 0.0f;       // per-query running sum

  const int kc_m = tid & 15, kc_c = tid >> 4;  // K staging coords
  const int vc_o = tid & 31, vc_m = tid >> 5;  // V staging coords

  for (int m0 = 0; m0 < NN; m0 += 16) {
    __syncthreads();
#pragma unroll
    for (int i = 0; i < 4; ++i)
      ldsK[kc_m * 66 + kc_c + 16 * i] =
          xb[(size_t)(kc_c + 16 * i) * NN + m0 + kc_m];
#pragma unroll
    for (int i = 0; i < 2; ++i)
      ldsV[vc_o * 18 + vc_m + 8 * i] =
          vb[(size_t)(m0 + vc_m + 8 * i) * C2 + vc_o];
    __syncthreads();

    // S^T(16 keys x 16 queries) = K_tile(16x64) * Q^T(64x16), K-dim chunks of 4
    v8f s = {};
    const int arow = lq * 66 + (lo ? 0 : 2);
#pragma unroll
    for (int ck = 0; ck < 16; ++ck) {
      v2f a;
      a[0] = ldsK[arow + 4 * ck];
      a[1] = ldsK[arow + 4 * ck + 1];
      s = wmma4(a, qb[ck], s);
    }

    // online softmax: key axis lives inside the lane (8 regs + half-swap)
    float tmax = s[0];
#pragma unroll
    for (int r = 1; r < 8; ++r) tmax = fmaxf(tmax, s[r]);
    tmax = fmaxf(tmax, sw16(tmax));
    float m_new = fmaxf(m_run, tmax);
    float scale = __expf(m_run - m_new);
    m_run = m_new;
    v8f p;
    float tsum = 0.0f;
#pragma unroll
    for (int r = 0; r < 8; ++r) {
      p[r] = __expf(s[r] - m_new);
      tsum += p[r];
    }
    tsum += sw16(tsum);
    l_run = l_run * scale + tsum;
#pragma unroll
    for (int r = 0; r < 8; ++r) {
      acc0[r] *= scale;
      acc1[r] *= scale;
    }

    // P^T (C layout) -> B fragments
    v2f pb[4];
    c2b(p, lo, pb);

    // y^T += V^T(32x16) * P^T(16x16)
    const int vrow0 = lq * 18 + (lo ? 0 : 2);
    const int vrow1 = (lq + 16) * 18 + (lo ? 0 : 2);
#pragma unroll
    for (int j = 0; j < 4; ++j) {
      v2f a;
      a[0] = ldsV[vrow0 + 4 * j];
      a[1] = ldsV[vrow0 + 4 * j + 1];
      acc0 = wmma4(a, pb[j], acc0);
    }
#pragma unroll
    for (int j = 0; j < 4; ++j) {
      v2f a;
      a[0] = ldsV[vrow1 + 4 * j];
      a[1] = ldsV[vrow1 + 4 * j + 1];
      acc1 = wmma4(a, pb[j], acc1);
    }
  }

  // normalize: y^T /= l   (per-lane, mirrored halves)
  float rinv = 1.0f / l_run;
#pragma unroll
  for (int r = 0; r < 8; ++r) {
    acc0[r] *= rinv;
    acc1[r] *= rinv;
  }

  // y^T -> B fragments over the o axis (K = 32 -> 8 chunks)
  v2f yb[8];
  c2b(acc0, lo, yb);
  c2b(acc1, lo, yb + 4);

  // z^T(c x q) = W_w(64x32) * y^T(32x16), bias + residual folded into C
  float* ob = out + (size_t)b * (CC * NN);
#pragma unroll
  for (int ct = 0; ct < 4; ++ct) {
    v8f z;
#pragma unroll
    for (int r = 0; r < 8; ++r) {
      int c = ct * 16 + r + (lo ? 0 : 8);
      z[r] = xb[(size_t)c * NN + n0 + lq] + W_b[c];
    }
    const int wrow = (ct * 16 + lq) * C2 + (lo ? 0 : 2);
#pragma unroll
    for (int oc = 0; oc < 8; ++oc) {
      v2f a;
      a[0] = W_w[wrow + 4 * oc];
      a[1] = W_w[wrow + 4 * oc + 1];
      z = wmma4(a, yb[oc], z);
    }
#pragma unroll
    for (int r = 0; r < 8; ++r) {
      int c = ct * 16 + r + (lo ? 0 : 8);
      ob[(size_t)c * NN + n0 + lq] = z[r];
    }
  }
}

extern "C" void kernel_launch(void* const* d_in, const int* in_sizes, int n_in,
                              void* d_out, int out_size, void* d_ws,
                              size_t ws_size, hipStream_t stream) {
  const float* x   = (const float*)d_in[0];
  const float* g_w = (const float*)d_in[1];
  const float* g_b = (const float*)d_in[2];
  const float* W_w = (const float*)d_in[3];
  const float* W_b = (const float*)d_in[4];
  float* out = (float*)d_out;
  float* Vg  = (float*)d_ws;  // [8][4096][32] f32 = 4 MB

  gx_kernel<<<(NB * NN) / 256, 256, 0, stream>>>(x, g_w, g_b, Vg);
  attn_kernel<<<NB * (NN / 128), 256, 0, stream>>>(x, Vg, W_w, W_b, out);
}